// CrossLAL_43482248904967
// MI455X (gfx1250) — compile-verified
//
#include <hip/hip_runtime.h>

#define B_    8
#define NQ_   4096
#define NKV_  4096
#define C_    256
#define K_    16
#define H_    32
#define EPS_  1e-5f
#define SLOPE_ 0.2f

typedef __attribute__((ext_vector_type(2))) float v2f;
typedef __attribute__((ext_vector_type(8))) float v8f;

// ---------------------------------------------------------------------------
// Kernel 0: repack weights once:  wbuf[0][h][c] = w1a[h][c]
//                                 wbuf[1][h][c] = w1b[h][c] - w1a[h][c]
// Removes all branching / on-the-fly differencing from the proj GEMM loop.
// ---------------------------------------------------------------------------
__global__ __launch_bounds__(256) void prep_w_kernel(const float* __restrict__ w1,
                                                     float* __restrict__ wbuf) {
    const int i = blockIdx.x * 256 + threadIdx.x;       // 0 .. H_*C_-1
    const int h = i >> 8;
    const int c = i & 255;
    const float a  = w1[(size_t)h * 512 + c];
    const float bb = w1[(size_t)h * 512 + 256 + c];
    wbuf[i]             = a;                            // w1a
    wbuf[H_ * C_ + i]   = bb - a;                       // w1d
}

// ---------------------------------------------------------------------------
// Kernel 1: projKV = kvf @ w1a^T ; projQ = qf @ w1d^T   (f32 WMMA 16x16x4)
// One wave = 16 rows x 32 cols (two 16-col tiles share the A fragment).
// A layout (16x4 f32): lane m = lane&15, K = 2*half + vgpr_j.
// B layout (4x16):     lane n = lane&15, K = 2*half + vgpr_j.
// D layout (16x16):    vgpr j -> M = j + 8*half, N = lane&15.
// ---------------------------------------------------------------------------
__global__ __launch_bounds__(256) void proj_kernel(const float* __restrict__ feat_kv,
                                                   const float* __restrict__ feat_q,
                                                   const float* __restrict__ wbuf,
                                                   float* __restrict__ projKV,
                                                   float* __restrict__ projQ) {
    const int which = blockIdx.y;                       // 0 = KV, 1 = Q (uniform)
    const float* feat = which ? feat_q : feat_kv;
    const float* wsel = wbuf + (size_t)which * (H_ * C_);
    float* dst        = which ? projQ  : projKV;

    const int wave = threadIdx.x >> 5;
    const int lane = threadIdx.x & 31;
    const int n    = lane & 15;
    const int half = lane >> 4;
    const int rbase = blockIdx.x * 128 + wave * 16;     // 32768 rows total
    const float* frow = feat + (size_t)(rbase + n) * C_;
    const float* wr0  = wsel + (size_t)n * C_;          // h = n
    const float* wr1  = wsel + (size_t)(n + 16) * C_;   // h = n + 16

    v8f acc0 = {}; v8f acc1 = {};
    #pragma unroll 8
    for (int kb = 0; kb < C_; kb += 4) {
        const int kk = kb + 2 * half;
        const v2f a  = *(const v2f*)&frow[kk];
        const v2f b0 = *(const v2f*)&wr0[kk];
        const v2f b1 = *(const v2f*)&wr1[kk];
        acc0 = __builtin_amdgcn_wmma_f32_16x16x4_f32(false, a, false, b0, (short)0, acc0, false, false);
        acc1 = __builtin_amdgcn_wmma_f32_16x16x4_f32(false, a, false, b1, (short)0, acc1, false, false);
    }
    #pragma unroll
    for (int j = 0; j < 8; ++j) {
        const int r = rbase + j + 8 * half;
        dst[(size_t)r * H_ + n]      = acc0[j];
        dst[(size_t)r * H_ + 16 + n] = acc1[j];
    }
}

// ---------------------------------------------------------------------------
// Kernel 2: exact squared distances via one f32 WMMA per 16q x 16kv tile
// (A row = (kx,ky,kz,|k|^2), B col = (-2qx,-2qy,-2qz,1), C = |q|^2),
// then per-lane register top-16 + cross-half merge via __shfl_xor(.,16).
// ---------------------------------------------------------------------------
__global__ __launch_bounds__(256) void topk_kernel(const float* __restrict__ qxyz,
                                                   const float* __restrict__ kxyz,
                                                   int* __restrict__ topk) {
    __shared__ float4 skv[NKV_];                        // 64 KB: kv xyz + |k|^2
    const int b   = blockIdx.y;
    const int tid = threadIdx.x;
    for (int i = tid; i < NKV_; i += 256) {
        const size_t o = ((size_t)b * NKV_ + i) * 3;
        const float x = kxyz[o], y = kxyz[o + 1], z = kxyz[o + 2];
        skv[i] = make_float4(x, y, z, x * x + y * y + z * z);
    }
    __syncthreads();

    const int wave = tid >> 5;
    const int lane = tid & 31;
    const int n    = lane & 15;
    const int half = lane >> 4;
    const int q    = blockIdx.x * 128 + wave * 16 + n;
    const size_t qo = ((size_t)b * NQ_ + q) * 3;
    const float qx = qxyz[qo], qy = qxyz[qo + 1], qz = qxyz[qo + 2];
    const float qn2 = qx * qx + qy * qy + qz * qz;

    v2f bf;                                             // B frag: constant over chunks
    if (half == 0) { bf.x = -2.f * qx; bf.y = -2.f * qy; }
    else           { bf.x = -2.f * qz; bf.y = 1.f;      }

    float db[16]; int ib[16];
    #pragma unroll
    for (int i = 0; i < 16; ++i) { db[i] = 3.0e38f; ib[i] = 0; }

    for (int chunk = 0; chunk < NKV_ / 16; ++chunk) {
        const float4 s = skv[chunk * 16 + n];
        v2f af;
        if (half == 0) { af.x = s.x; af.y = s.y; }
        else           { af.x = s.z; af.y = s.w; }
        v8f c;
        #pragma unroll
        for (int j = 0; j < 8; ++j) c[j] = qn2;
        c = __builtin_amdgcn_wmma_f32_16x16x4_f32(false, af, false, bf, (short)0, c, false, false);

        #pragma unroll
        for (int j = 0; j < 8; ++j) {
            const float d  = c[j];
            const int kvid = chunk * 16 + j + 8 * half; // D: M = j + 8*half
            if (d < db[15]) {                           // sorted-ascending insert
                float dv = d; int iv = kvid;
                #pragma unroll
                for (int i = 0; i < 16; ++i) {
                    const bool sm = dv < db[i];
                    const float td = db[i]; const int ti = ib[i];
                    db[i] = sm ? dv : td;  ib[i] = sm ? iv : ti;
                    dv    = sm ? td : dv;  iv    = sm ? ti : iv;
                }
            }
        }
    }

    // merge lane l with lane l+16 (disjoint kv subsets of the same query):
    // snapshot partner's lists first, then insert (both halves mutate).
    float pd[16]; int pi[16];
    #pragma unroll
    for (int e = 0; e < 16; ++e) {
        pd[e] = __shfl_xor(db[e], 16, 32);
        pi[e] = __shfl_xor(ib[e], 16, 32);
    }
    #pragma unroll
    for (int e = 0; e < 16; ++e) {
        float dv = pd[e]; int iv = pi[e];
        if (dv < db[15]) {
            #pragma unroll
            for (int i = 0; i < 16; ++i) {
                const bool sm = dv < db[i];
                const float td = db[i]; const int ti = ib[i];
                db[i] = sm ? dv : td;  ib[i] = sm ? iv : ti;
                dv    = sm ? td : dv;  iv    = sm ? ti : iv;
            }
        }
    }
    if (half == 0) {
        #pragma unroll
        for (int k = 0; k < K_; ++k)
            topk[((size_t)b * NQ_ + q) * K_ + k] = ib[k];
    }
}

// ---------------------------------------------------------------------------
// Kernel 3: x[q,k,h] = projKV[idx,h] + projQ[q,h]; affine; leaky; max over k.
// One wave per query, lane == h (H == 32 == wave32). Gathers are 128B/row.
// ---------------------------------------------------------------------------
__global__ __launch_bounds__(256) void gathermax_kernel(const float* __restrict__ projKV,
                                                        const float* __restrict__ projQ,
                                                        const int* __restrict__ topk,
                                                        const float* __restrict__ g1,
                                                        const float* __restrict__ b1,
                                                        const float* __restrict__ m1,
                                                        const float* __restrict__ v1,
                                                        float* __restrict__ xmax) {
    const int wave = threadIdx.x >> 5;
    const int lane = threadIdx.x & 31;
    const size_t qg = (size_t)blockIdx.x * 8 + wave;    // row over B*NQ
    const int b = (int)(qg >> 12);                      // / 4096
    const float s  = g1[lane] * rsqrtf(v1[lane] + EPS_);
    const float bb = b1[lane] - m1[lane] * s;
    const float pq = projQ[qg * H_ + lane];
    float acc = -3.0e38f;
    #pragma unroll
    for (int k = 0; k < K_; ++k) {
        const int idx = topk[qg * K_ + k];
        float t = (projKV[((size_t)b * NKV_ + idx) * H_ + lane] + pq) * s + bb;
        t = t >= 0.f ? t : SLOPE_ * t;
        acc = fmaxf(acc, t);
    }
    xmax[qg * H_ + lane] = acc;
}

// ---------------------------------------------------------------------------
// Kernel 4: y = xmax (32768x32) @ w2^T (32->256), affine + leaky, via f32 WMMA.
// One wave = 16 rows x 16 cols, K = 32 in 8 WMMA steps.
// ---------------------------------------------------------------------------
__global__ __launch_bounds__(256) void out_kernel(const float* __restrict__ xmax,
                                                  const float* __restrict__ w2,
                                                  const float* __restrict__ g2,
                                                  const float* __restrict__ b2,
                                                  const float* __restrict__ m2,
                                                  const float* __restrict__ v2,
                                                  float* __restrict__ out) {
    const int wave = threadIdx.x >> 5;
    const int lane = threadIdx.x & 31;
    const int n    = lane & 15;
    const int half = lane >> 4;
    const int rbase = blockIdx.x * 16;
    const int col   = blockIdx.y * 128 + wave * 16 + n;
    const float* arow = xmax + (size_t)(rbase + n) * H_;
    const float* wrow = w2 + (size_t)col * H_;
    v8f acc = {};
    #pragma unroll
    for (int kb = 0; kb < H_; kb += 4) {
        const int kk = kb + 2 * half;
        const v2f a  = *(const v2f*)&arow[kk];
        const v2f bm = *(const v2f*)&wrow[kk];
        acc = __builtin_amdgcn_wmma_f32_16x16x4_f32(false, a, false, bm, (short)0, acc, false, false);
    }
    const float s  = g2[col] * rsqrtf(v2[col] + EPS_);
    const float bb = b2[col] - m2[col] * s;
    #pragma unroll
    for (int j = 0; j < 8; ++j) {
        float t = acc[j] * s + bb;
        t = t >= 0.f ? t : SLOPE_ * t;
        out[(size_t)(rbase + j + 8 * half) * C_ + col] = t;
    }
}

// ---------------------------------------------------------------------------
extern "C" void kernel_launch(void* const* d_in, const int* in_sizes, int n_in,
                              void* d_out, int out_size, void* d_ws, size_t ws_size,
                              hipStream_t stream) {
    const float* qf   = (const float*)d_in[0];
    const float* qxyz = (const float*)d_in[1];
    const float* kvf  = (const float*)d_in[2];
    const float* kxyz = (const float*)d_in[3];
    const float* w1   = (const float*)d_in[4];
    const float* g1   = (const float*)d_in[5];
    const float* b1   = (const float*)d_in[6];
    const float* m1   = (const float*)d_in[7];
    const float* v1   = (const float*)d_in[8];
    const float* w2   = (const float*)d_in[9];
    const float* g2   = (const float*)d_in[10];
    const float* b2   = (const float*)d_in[11];
    const float* m2   = (const float*)d_in[12];
    const float* v2   = (const float*)d_in[13];
    float* out = (float*)d_out;

    float* projKV = (float*)d_ws;                            // 4 MB
    float* projQ  = projKV + (size_t)B_ * NKV_ * H_;         // 4 MB
    float* xmax   = projQ  + (size_t)B_ * NQ_  * H_;         // 4 MB
    int*   topk   = (int*)(xmax + (size_t)B_ * NQ_ * H_);    // 2 MB
    float* wbuf   = (float*)(topk + (size_t)B_ * NQ_ * K_);  // 64 KB (w1a ; w1d)

    prep_w_kernel<<<(H_ * C_) / 256, 256, 0, stream>>>(w1, wbuf);
    proj_kernel<<<dim3((B_ * NKV_) / 128, 2), 256, 0, stream>>>(kvf, qf, wbuf, projKV, projQ);
    topk_kernel<<<dim3(NQ_ / 128, B_), 256, 0, stream>>>(qxyz, kxyz, topk);
    gathermax_kernel<<<(B_ * NQ_) / 8, 256, 0, stream>>>(projKV, projQ, topk, g1, b1, m1, v1, xmax);
    out_kernel<<<dim3((B_ * NQ_) / 16, 2), 256, 0, stream>>>(xmax, w2, g2, b2, m2, v2, out);
}